// LightConv_38311108280984
// MI455X (gfx1250) — compile-verified
//
#include <hip/hip_runtime.h>
#include <cstdint>

#ifndef __has_builtin
#define __has_builtin(x) 0
#endif

#if __has_builtin(__builtin_amdgcn_global_load_async_to_lds_b32)
#define HAVE_ASYNC_LDS 1
#else
#define HAVE_ASYNC_LDS 0
#endif

#define THREADS 256

// ---- address-space cast helpers (generic -> LDS(3)/global(1)) --------------
// Generic pointers to __shared__ carry the LDS byte offset in the low 32 bits
// (truncating to 32 bits yields it); integer round-trip moves between address
// spaces. The async builtin takes typed, non-const AS pointers.
typedef __attribute__((address_space(1))) int   gint_t;
typedef __attribute__((address_space(3))) int   lint_t;
typedef __attribute__((address_space(1))) float gfloat_t;

__device__ __forceinline__ lint_t* to_lds_i32(void* p) {
    return (lint_t*)(uint32_t)(uint64_t)p;
}
__device__ __forceinline__ gint_t* to_gbl_i32(const void* p) {
    return (gint_t*)(uint64_t)p;   // const shed intentionally; op only reads
}

__device__ __forceinline__ void atomic_add_f32(float* addr, float v) {
#if __has_builtin(__builtin_amdgcn_global_atomic_fadd_f32)
    // Guaranteed native global_atomic_add_f32; unused result -> no-return
    // encoding (STOREcnt-tracked, no CAS loop).
    (void)__builtin_amdgcn_global_atomic_fadd_f32((gfloat_t*)(uint64_t)addr, v);
#else
    __hip_atomic_fetch_add(addr, v, __ATOMIC_RELAXED, __HIP_MEMORY_SCOPE_AGENT);
#endif
}

// ---- zeroing ---------------------------------------------------------------
__global__ void zero_f4_kernel(float4* __restrict__ p, long n4) {
    long i = (long)blockIdx.x * blockDim.x + threadIdx.x;
    long stride = (long)gridDim.x * blockDim.x;
    float4 z = make_float4(0.f, 0.f, 0.f, 0.f);
    for (; i < n4; i += stride) p[i] = z;
}

__global__ void zero_f1_kernel(float* __restrict__ p, int n) {
    int i = blockIdx.x * blockDim.x + threadIdx.x;
    int stride = gridDim.x * blockDim.x;
    for (; i < n; i += stride) p[i] = 0.f;
}

// ---- out-degree histogram --------------------------------------------------
__global__ void degree_kernel(const int* __restrict__ src,
                              float* __restrict__ deg, int n_edges) {
    int i = blockIdx.x * blockDim.x + threadIdx.x;
    int stride = gridDim.x * blockDim.x;
    for (; i < n_edges; i += stride)
        atomic_add_f32(&deg[src[i]], 1.0f);
}

// ---- norm = deg > 0 ? rsqrt(deg) : 0  (in place) --------------------------
__global__ void norm_kernel(float* __restrict__ deg_norm, int n_nodes) {
    int i = blockIdx.x * blockDim.x + threadIdx.x;
    if (i < n_nodes) {
        float d = deg_norm[i];
        deg_norm[i] = (d > 0.f) ? rsqrtf(d) : 0.f;
    }
}

// ---- main scatter: out[dst,:] += norm[src]*norm[dst]*feat[src,:] ----------
// One 256-thread block = 8 wave32s handles a tile of 256 edges.
// Edge indices are streamed into LDS via gfx1250 async-to-LDS; each wave then
// processes 32 edges, with lane == feature column (D_FEAT == 32 == wave32).
__global__ __launch_bounds__(THREADS)
void scatter_kernel(const float* __restrict__ feat,
                    const int* __restrict__ src,
                    const int* __restrict__ dst,
                    const float* __restrict__ norm,
                    float* __restrict__ out, int n_edges) {
    __shared__ int   ls[THREADS];
    __shared__ int   ld[THREADS];
    __shared__ float lw[THREADS];

    const int tid  = threadIdx.x;
    const int lane = tid & 31;
    const int wv   = tid >> 5;

    const long base = (long)blockIdx.x * THREADS;
    if (base >= (long)n_edges) return;
    const long rem   = (long)n_edges - base;
    const int  valid = (rem >= THREADS) ? THREADS : (int)rem;

    if (valid == THREADS) {
#if HAVE_ASYNC_LDS
        // gfx1250 async DMA: global -> LDS, tracked on ASYNCcnt
        __builtin_amdgcn_global_load_async_to_lds_b32(
            to_gbl_i32(src + base + tid), to_lds_i32(&ls[tid]), 0, 0);
        __builtin_amdgcn_global_load_async_to_lds_b32(
            to_gbl_i32(dst + base + tid), to_lds_i32(&ld[tid]), 0, 0);
#if __has_builtin(__builtin_amdgcn_s_wait_asynccnt)
        __builtin_amdgcn_s_wait_asynccnt(0);
#else
        asm volatile("s_wait_asynccnt 0" ::: "memory");
#endif
#else
        ls[tid] = src[base + tid];
        ld[tid] = dst[base + tid];
#endif
    } else if (tid < valid) {
        ls[tid] = src[base + tid];
        ld[tid] = dst[base + tid];
    }
    __syncthreads();

    // per-edge weight, computed once by 256 threads in parallel
    if (tid < valid) lw[tid] = norm[ls[tid]] * norm[ld[tid]];
    __syncthreads();

    const int ebeg = wv * 32;
    const int eend = (ebeg + 32 < valid) ? (ebeg + 32) : valid;
    for (int i = ebeg; i < eend; ++i) {
        const int   s = ls[i];
        const int   d = ld[i];
        const float w = lw[i];
        if (i + 4 < eend) {
            // global_prefetch_b8: pull a future random feature row toward cache
            __builtin_prefetch(&feat[(long)ls[i + 4] * 32 + lane], 0, 0);
        }
        const float v = feat[(long)s * 32 + lane] * w;   // coalesced 128B/wave
        atomic_add_f32(&out[(long)d * 32 + lane], v);    // coalesced 128B/wave
    }
}

// ---------------------------------------------------------------------------
extern "C" void kernel_launch(void* const* d_in, const int* in_sizes, int n_in,
                              void* d_out, int out_size, void* d_ws, size_t ws_size,
                              hipStream_t stream) {
    const float* feat = (const float*)d_in[0];
    const int*   src  = (const int*)d_in[1];
    const int*   dst  = (const int*)d_in[2];
    float*       out  = (float*)d_out;

    const int n_edges = in_sizes[1];
    const int n_nodes = in_sizes[0] / 32;   // D_FEAT = 32

    float* deg_norm = (float*)d_ws;         // n_nodes floats; deg -> norm in place

    // 1) zero output (out_size = n_nodes*32, multiple of 4)
    {
        long n4 = (long)out_size / 4;
        long b  = (n4 + 255) / 256;
        int blocks = (b > 2048) ? 2048 : (int)b;
        zero_f4_kernel<<<blocks, 256, 0, stream>>>((float4*)out, n4);
    }
    // 2) zero degree accumulator
    {
        int b = (n_nodes + 255) / 256;
        int blocks = (b > 1024) ? 1024 : b;
        zero_f1_kernel<<<blocks, 256, 0, stream>>>(deg_norm, n_nodes);
    }
    // 3) out-degree histogram
    {
        int b = (n_edges + 255) / 256;
        int blocks = (b > 4096) ? 4096 : b;
        degree_kernel<<<blocks, 256, 0, stream>>>(src, deg_norm, n_edges);
    }
    // 4) deg -> rsqrt norm
    {
        int blocks = (n_nodes + 255) / 256;
        norm_kernel<<<blocks, 256, 0, stream>>>(deg_norm, n_nodes);
    }
    // 5) fused normalized scatter-add (one tile of 256 edges per block)
    {
        long blocks = ((long)n_edges + THREADS - 1) / THREADS;
        scatter_kernel<<<(int)blocks, THREADS, 0, stream>>>(
            feat, src, dst, deg_norm, out, n_edges);
    }
}